// DynamicsBasic_53498112639162
// MI455X (gfx1250) — compile-verified
//
#include <hip/hip_runtime.h>

// CDNA5 / gfx1250. wave32. Fused dual-GEMV via V_WMMA_F32_16X16X4_F32.
//
// out[m] = s*(1-s) * (W @ (x_f - x_i))[m],  s = sigmoid((W @ x_i)[m] + b[m])
//
// Memory-bound: streams W (256 MB) exactly once; WMMA does the K-reduction.

typedef __attribute__((ext_vector_type(2))) float v2f;
typedef __attribute__((ext_vector_type(8))) float v8f;

#if defined(__HIP_DEVICE_COMPILE__)
#if !__has_builtin(__builtin_amdgcn_wmma_f32_16x16x4_f32)
#error "missing __builtin_amdgcn_wmma_f32_16x16x4_f32 on this toolchain"
#endif
#endif

#define CHUNK 1024          // K elements of x_i / v staged in LDS per pass
#define WAVES_PER_BLOCK 4   // 4 waves * 16 rows = 64 rows per block

// Kernel 1: each wave owns a 16-row block of W and one K-segment.
// Accumulates C[16x16] where column 0 = partial W@x_i, column 1 = partial W@v.
__global__ __launch_bounds__(WAVES_PER_BLOCK * 32)
void dyn_gemv2_wmma(const float* __restrict__ W,
                    const float* __restrict__ x_i,
                    const float* __restrict__ x_f,
                    float2* __restrict__ ws,   // [M][ks] partials {Wx, Wv}
                    int D, int ks)
{
    __shared__ float s_xi[CHUNK];
    __shared__ float s_v [CHUNK];

    const int tid  = threadIdx.x;
    const int lane = tid & 31;
    const int wave = tid >> 5;
    const int l15  = lane & 15;
    const int half = lane >> 4;            // 0: K pair (0,1); 1: K pair (2,3)
    const int kseg = blockIdx.y;
    const int segLen  = D / ks;
    const int segBase = kseg * segLen;

    const int mBase = (blockIdx.x * WAVES_PER_BLOCK + wave) * 16;
    // A-matrix: lane holds W[mBase + l15][k + 2*half .. +1]
    const float* __restrict__ Wrow = W + (size_t)(mBase + l15) * (size_t)D + 2 * half;

    // B-matrix lane role: column N = l15. col0 = x_i, col1 = v, others 0.
    const float bmask = (l15 <= 1) ? 1.0f : 0.0f;

    v8f c = {0.f, 0.f, 0.f, 0.f, 0.f, 0.f, 0.f, 0.f};

    for (int kc = segBase; kc < segBase + segLen; kc += CHUNK) {
        __syncthreads();
        // Cooperative stage of x_i and v = x_f - x_i for this chunk.
        for (int i = tid * 2; i < CHUNK; i += blockDim.x * 2) {
            v2f xi = *(const v2f*)(x_i + kc + i);
            v2f xf = *(const v2f*)(x_f + kc + i);
            *(v2f*)(s_xi + i) = xi;
            v2f vv; vv.x = xf.x - xi.x; vv.y = xf.y - xi.y;
            *(v2f*)(s_v + i) = vv;
        }
        __syncthreads();

        const float* __restrict__ bsrc = (l15 == 1) ? s_v : s_xi;

        #pragma unroll 4
        for (int kk = 0; kk < CHUNK; kk += 4) {
            // A tile 16x4: one b64 per lane, sequential along each W row.
            v2f a = *(const v2f*)(Wrow + kc + kk);
            // B tile 4x16: pair at K = kk + 2*half of this lane's column.
            v2f p = *(const v2f*)(bsrc + kk + 2 * half);
            v2f bb; bb.x = p.x * bmask; bb.y = p.y * bmask;
            c = __builtin_amdgcn_wmma_f32_16x16x4_f32(
                    false, a, false, bb, (short)0, c, false, false);
        }
    }

    // C layout: VGPR r, lanes 0-15 -> (M=r, N=lane); lanes 16-31 -> (M=r+8).
    // Wx lives in lanes 0/16 (N=0), Wv in lanes 1/17 (N=1).
    union { v8f v; float f[8]; } cu; cu.v = c;
    float wv[8];
    #pragma unroll
    for (int r = 0; r < 8; ++r)
        wv[r] = __shfl_xor(cu.f[r], 1, 32);   // pull N=1 next to N=0

    if (l15 == 0) {
        const int row0 = mBase + 8 * half;
        #pragma unroll
        for (int r = 0; r < 8; ++r) {
            float2 pr; pr.x = cu.f[r]; pr.y = wv[r];
            ws[(size_t)(row0 + r) * ks + kseg] = pr;
        }
    }
}

// Kernel 2: reduce K-split partials, add bias, apply sigmoid-derivative gate.
__global__ __launch_bounds__(256)
void dyn_finalize(const float2* __restrict__ ws,
                  const float* __restrict__ bias,
                  float* __restrict__ out, int M, int ks)
{
    int m = blockIdx.x * blockDim.x + threadIdx.x;
    if (m >= M) return;
    float wx = 0.f, wv = 0.f;
    for (int j = 0; j < ks; ++j) {
        float2 p = ws[(size_t)m * ks + j];
        wx += p.x; wv += p.y;
    }
    float a = wx + bias[m];
    float s = 1.0f / (1.0f + __expf(-a));
    out[m] = s * (1.0f - s) * wv;
}

extern "C" void kernel_launch(void* const* d_in, const int* in_sizes, int n_in,
                              void* d_out, int out_size, void* d_ws, size_t ws_size,
                              hipStream_t stream) {
    // setup_inputs order: t(1), y(M), x_i(D), x_f(D), W(M*D), b(M). t,y unused.
    const float* x_i  = (const float*)d_in[2];
    const float* x_f  = (const float*)d_in[3];
    const float* W    = (const float*)d_in[4];
    const float* bias = (const float*)d_in[5];
    float* out = (float*)d_out;

    const int D = in_sizes[2];
    const int M = in_sizes[5];

    // K-split for occupancy; clamp to workspace capacity and CHUNK divisibility.
    int ks = 8;
    while (ks > 1 && (size_t)M * (size_t)ks * sizeof(float2) > ws_size) ks >>= 1;
    while (ks > 1 && ((D / ks) % CHUNK) != 0) ks >>= 1;

    float2* ws = (float2*)d_ws;

    dim3 grid1(M / (WAVES_PER_BLOCK * 16), ks);
    dyn_gemv2_wmma<<<grid1, WAVES_PER_BLOCK * 32, 0, stream>>>(W, x_i, x_f, ws, D, ks);

    dyn_finalize<<<(M + 255) / 256, 256, 0, stream>>>(ws, bias, out, M, ks);
}